// LatentActionGen_27857157881910
// MI455X (gfx1250) — compile-verified
//
#include <hip/hip_runtime.h>
#include <math.h>

typedef float v2f __attribute__((ext_vector_type(2)));
typedef float v8f __attribute__((ext_vector_type(8)));

// Problem constants
#define MROWS 18432       // 512 * 36  (N*H*W)
#define NB    512
#define CIN   128
#define EMB   128
#define KCODE 512
#define EPSF  1e-5f

// ---------------------------------------------------------------------------
// Layout transforms
// ---------------------------------------------------------------------------
__global__ void k_nchw2nhwc(const float* __restrict__ in, float* __restrict__ out) {
  int i = blockIdx.x * blockDim.x + threadIdx.x;        // over 512*128*36
  if (i >= NB * CIN * 36) return;
  int hw = i % 36; int r = i / 36; int c = r % CIN; int n = r / CIN;
  out[((size_t)(n * 36 + hw)) * CIN + c] = in[i];
}

// W[Cout][128][3][3] (OIHW) -> Wtp[9][64][Cout][2]   (k-pairs interleaved so a
// B fragment for WMMA K-step is one b64 load per lane)
__global__ void k_transpose_wp(const float* __restrict__ W, float* __restrict__ Wtp, int Cout) {
  int i = blockIdx.x * blockDim.x + threadIdx.x;
  int total = 9 * CIN * Cout;
  if (i >= total) return;
  int lo = i & 1; int t = i >> 1;
  int co = t % Cout; t /= Cout;
  int k2 = t % 64;  int tap = t / 64;
  int k = k2 * 2 + lo;
  Wtp[i] = W[((size_t)co * CIN + k) * 9 + tap];
}

__global__ void k_zero(float* __restrict__ p, int n) {
  int i = blockIdx.x * blockDim.x + threadIdx.x;
  if (i < n) p[i] = 0.0f;
}

// ---------------------------------------------------------------------------
// Implicit-GEMM 3x3 conv, pad 1, stride 1, via V_WMMA_F32_16X16X4_F32.
// X: [MROWS,128] NHWC-flat, Wtp: [9][64][COUT][2], Y: [MROWS,COUT]
// Workgroup = 32-row M tile; wave w owns C_out columns [16w,16w+16) and BOTH
// 16-row halves (two accumulators) so each B fragment is reused by 2 WMMAs.
// A tile (32x128, 16 KB) staged in LDS once per tap, shared by all waves.
// FLAGS: bit0 = accumulate into Y, bit1 = relu   (compile-time)
// ---------------------------------------------------------------------------
template <int COUT, int FLAGS>
__global__ void __launch_bounds__(256) k_conv3_wmma(
    const float* __restrict__ X, const float* __restrict__ Wtp,
    float* __restrict__ Y) {
  __shared__ float As[32 * 128];                        // 16 KB
  const int tid   = threadIdx.x;
  const int lane  = tid & 31;
  const int hi    = lane >> 4;                          // K-pair select
  const int nT    = lane & 15;
  const int wave  = tid >> 5;
  const int mbase = blockIdx.x * 32;
  const int co    = wave * 16 + nT;

  v8f acc0 = {};
  v8f acc1 = {};
  for (int tap = 0; tap < 9; ++tap) {
    const int dh = tap / 3 - 1, dw = tap % 3 - 1;
    __syncthreads();
    // stage shifted 32x128 A tile (zeros outside the 6x6 image)
    for (int i = tid; i < 1024; i += blockDim.x) {      // 1024 float4
      int r  = i >> 5;
      int k4 = (i & 31) << 2;
      int mrow = mbase + r;
      int n = mrow / 36, hw = mrow % 36;
      int h2 = hw / 6 + dh, w2 = hw % 6 + dw;
      float4 v = {0.f, 0.f, 0.f, 0.f};
      if ((unsigned)h2 < 6u && (unsigned)w2 < 6u)
        v = *(const float4*)(X + (size_t)(n * 36 + h2 * 6 + w2) * CIN + k4);
      *(float4*)(&As[r * 128 + k4]) = v;
    }
    __syncthreads();
    // B base for this tap / column; pairs interleaved -> one b64 per K-step
    const float* bp  = Wtp + ((size_t)tap * 64 * COUT + co) * 2;
    const float* ap0 = &As[nT * 128];
    const float* ap1 = &As[(nT + 16) * 128];
#pragma unroll 8
    for (int k0 = 0; k0 < 128; k0 += 4) {
      int ks = k0 + hi * 2;                             // even
      v2f b  = *(const v2f*)(bp + (size_t)(ks >> 1) * COUT * 2);  // global b64
      v2f a0 = *(const v2f*)(ap0 + ks);                 // ds_load_b64
      v2f a1 = *(const v2f*)(ap1 + ks);                 // ds_load_b64
      acc0 = __builtin_amdgcn_wmma_f32_16x16x4_f32(false, a0, false, b,
                                                   (short)0, acc0, false, false);
      acc1 = __builtin_amdgcn_wmma_f32_16x16x4_f32(false, a1, false, b,
                                                   (short)0, acc1, false, false);
    }
  }
#pragma unroll
  for (int r2 = 0; r2 < 8; ++r2) {                      // D: VGPR r -> M = r + hi*8
    int m0 = mbase + r2 + hi * 8;                       // tile 0 rows
    int m1 = m0 + 16;                                   // tile 1 rows
    size_t o0 = (size_t)m0 * COUT + co;
    size_t o1 = (size_t)m1 * COUT + co;
    float v0 = acc0[r2], v1 = acc1[r2];
    if (FLAGS & 1) { v0 += Y[o0]; v1 += Y[o1]; }
    if (FLAGS & 2) { v0 = fmaxf(v0, 0.f); v1 = fmaxf(v1, 0.f); }
    Y[o0] = v0;
    Y[o1] = v1;
  }
}

// ---------------------------------------------------------------------------
// Per-channel sum / sumsq over M rows (biased var later). 144 blocks x 256.
// ---------------------------------------------------------------------------
__global__ void k_chan_stats(const float* __restrict__ X,
                             float* __restrict__ sum, float* __restrict__ ssq) {
  __shared__ float s1[256], s2[256];
  const int c  = threadIdx.x & 127;
  const int rl = threadIdx.x >> 7;                      // 0/1
  const int m0 = blockIdx.x * 128;
  float a = 0.f, b = 0.f;
  for (int r = rl; r < 128; r += 2) {
    float v = X[(size_t)(m0 + r) * CIN + c];
    a += v; b += v * v;
  }
  s1[threadIdx.x] = a; s2[threadIdx.x] = b;
  __syncthreads();
  if (rl == 0) {
    atomicAdd(&sum[c], s1[c] + s1[c + 128]);
    atomicAdd(&ssq[c], s2[c] + s2[c + 128]);
  }
}

// out = relu( (raw - mean)*rsqrt(var+eps)*scale + bias [+ resid] )
__global__ void k_bn_act(const float* __restrict__ raw,
                         const float* __restrict__ sum, const float* __restrict__ ssq,
                         const float* __restrict__ scale, const float* __restrict__ bias,
                         const float* __restrict__ resid, float* __restrict__ out) {
  int i = blockIdx.x * blockDim.x + threadIdx.x;
  if (i >= MROWS * CIN) return;
  int c = i & 127;
  const float inv = 1.0f / (float)MROWS;
  float mean = sum[c] * inv;
  float var  = ssq[c] * inv - mean * mean;
  float v = (raw[i] - mean) * rsqrtf(var + EPSF) * scale[c] + bias[c];
  if (resid) v += resid[i];
  out[i] = fmaxf(v, 0.f);
}

// ---------------------------------------------------------------------------
// FC: flat[b][e] = sum_k yo[(b*36 + k%36)*64 + k/36] * fc_w[e][k] + fc_b[e]
// M=512, N=128, K=2304 with V_WMMA_F32_16X16X4_F32. 32 blocks x 8 waves.
// ---------------------------------------------------------------------------
__global__ void __launch_bounds__(256) k_fc_wmma(
    const float* __restrict__ A, const float* __restrict__ Wf,
    const float* __restrict__ bias, float* __restrict__ flat) {
  const int tid = threadIdx.x;
  const int lane = tid & 31;
  const int hi = lane >> 4;
  const int nT = lane & 15;
  const int e  = (tid >> 5) * 16 + nT;
  const int mbase = blockIdx.x * 16;
  const int b = mbase + nT;
  const float* wp = Wf + (size_t)e * 2304;
  v8f acc = {};
  for (int k0 = 0; k0 < 2304; k0 += 4) {
    int i0 = k0 + hi * 2, i1 = i0 + 1;                  // i0 even -> 8B aligned
    v2f av;
    av[0] = A[(size_t)(b * 36 + i0 % 36) * 64 + i0 / 36];
    av[1] = A[(size_t)(b * 36 + i1 % 36) * 64 + i1 / 36];
    v2f bv = *(const v2f*)(wp + i0);                    // one global b64
    acc = __builtin_amdgcn_wmma_f32_16x16x4_f32(false, av, false, bv,
                                                (short)0, acc, false, false);
  }
#pragma unroll
  for (int r = 0; r < 8; ++r) {
    int m = mbase + r + hi * 8;
    flat[(size_t)m * EMB + e] = acc[r] + bias[e];
  }
}

// BatchNorm1d stats over batch: mean/var per feature (biased)
__global__ void k_feat_stats(const float* __restrict__ flat,
                             float* __restrict__ fmean, float* __restrict__ fvar) {
  __shared__ float s1[256], s2[256];
  int e = blockIdx.x;
  float a = 0.f, b = 0.f;
  for (int m = threadIdx.x; m < NB; m += 256) {
    float v = flat[(size_t)m * EMB + e];
    a += v; b += v * v;
  }
  s1[threadIdx.x] = a; s2[threadIdx.x] = b;
  __syncthreads();
  for (int s = 128; s > 0; s >>= 1) {
    if (threadIdx.x < s) { s1[threadIdx.x] += s1[threadIdx.x + s];
                           s2[threadIdx.x] += s2[threadIdx.x + s]; }
    __syncthreads();
  }
  if (threadIdx.x == 0) {
    float mean = s1[0] / (float)NB;
    fmean[e] = mean;
    fvar[e]  = s2[0] / (float)NB - mean * mean;
  }
}

__global__ void k_bn1d(const float* __restrict__ flat, const float* __restrict__ fmean,
                       const float* __restrict__ fvar, float* __restrict__ out) {
  int i = blockIdx.x * blockDim.x + threadIdx.x;
  if (i >= NB * EMB) return;
  int e = i & 127;
  out[i] = (flat[i] - fmean[e]) * rsqrtf(fvar[e] + EPSF);
}

__global__ void k_emb_sq(const float* __restrict__ emb, float* __restrict__ ksq) {
  int k = blockIdx.x * blockDim.x + threadIdx.x;
  if (k >= KCODE) return;
  float s = 0.f;
  for (int e = 0; e < EMB; ++e) { float v = emb[(size_t)k * EMB + e]; s += v * v; }
  ksq[k] = s;
}

// argmin_k ( ||e_k||^2 - 2 f.e_k )  (||f||^2 constant wrt k)
__global__ void k_vq_argmin(const float* __restrict__ flatN, const float* __restrict__ emb,
                            const float* __restrict__ ksq, int* __restrict__ idx,
                            float* __restrict__ counts, float* __restrict__ oidx) {
  __shared__ float f[EMB];
  __shared__ float bv[256];
  __shared__ int   bi[256];
  int b = blockIdx.x, tid = threadIdx.x;
  if (tid < EMB) f[tid] = flatN[(size_t)b * EMB + tid];
  __syncthreads();
  float best = 3.4e38f; int besti = 0;
  for (int k = tid; k < KCODE; k += 256) {
    const float* ek = emb + (size_t)k * EMB;
    float dot = 0.f;
    for (int e = 0; e < EMB; ++e) dot += f[e] * ek[e];
    float d = ksq[k] - 2.f * dot;
    if (d < best) { best = d; besti = k; }
  }
  bv[tid] = best; bi[tid] = besti;
  __syncthreads();
  for (int s = 128; s > 0; s >>= 1) {
    if (tid < s) {
      if (bv[tid + s] < bv[tid] ||
          (bv[tid + s] == bv[tid] && bi[tid + s] < bi[tid])) {
        bv[tid] = bv[tid + s]; bi[tid] = bi[tid + s];
      }
    }
    __syncthreads();
  }
  if (tid == 0) {
    idx[b] = bi[0];
    oidx[b] = (float)bi[0];
    atomicAdd(&counts[bi[0]], 1.0f);
  }
}

// q = emb[idx]; loss = 2*mean((q-flat)^2); z = broadcast q over 6x6
__global__ void k_vq_out(const float* __restrict__ flatN, const float* __restrict__ emb,
                         const int* __restrict__ idx, float* __restrict__ z,
                         float* __restrict__ loss) {
  __shared__ float red[EMB];
  int b = blockIdx.x, e = threadIdx.x;
  float q = emb[(size_t)idx[b] * EMB + e];
  float d = q - flatN[(size_t)b * EMB + e];
  red[e] = d * d;
  __syncthreads();
  for (int s = 64; s > 0; s >>= 1) {
    if (e < s) red[e] += red[e + s];
    __syncthreads();
  }
  if (e == 0) loss[b] = 2.0f * red[0] / (float)EMB;
  float* zb = z + ((size_t)b * EMB + e) * 36;
#pragma unroll
  for (int hw = 0; hw < 36; ++hw) zb[hw] = q;
}

__global__ void k_perp(const float* __restrict__ counts, float* __restrict__ out) {
  __shared__ float red[KCODE];
  int k = threadIdx.x;
  float p = counts[k] * (1.0f / (float)NB);
  red[k] = -p * logf(p + 1e-10f);
  __syncthreads();
  for (int s = 256; s > 0; s >>= 1) {
    if (k < s) red[k] += red[k + s];
    __syncthreads();
  }
  if (k == 0) out[0] = expf(red[0]);
}

// ---------------------------------------------------------------------------
extern "C" void kernel_launch(void* const* d_in, const int* in_sizes, int n_in,
                              void* d_out, int out_size, void* d_ws, size_t ws_size,
                              hipStream_t stream) {
  (void)in_sizes; (void)n_in; (void)out_size; (void)ws_size;
  const float* s0         = (const float*)d_in[0];
  const float* s1         = (const float*)d_in[1];
  const float* conv_w     = (const float*)d_in[2];
  const float* convs_w    = (const float*)d_in[3];
  const float* rb_c1_w    = (const float*)d_in[4];
  const float* rb_bn1_s   = (const float*)d_in[5];
  const float* rb_bn1_b   = (const float*)d_in[6];
  const float* rb_c2_w    = (const float*)d_in[7];
  const float* rb_bn2_s   = (const float*)d_in[8];
  const float* rb_bn2_b   = (const float*)d_in[9];
  const float* conv_out_w = (const float*)d_in[10];
  const float* fc_w       = (const float*)d_in[11];
  const float* fc_b       = (const float*)d_in[12];
  const float* emb        = (const float*)d_in[13];

  float* out  = (float*)d_out;            // [z | loss | perplexity | idx]
  float* z    = out;                      // 512*128*36
  float* loss = out + 2359296;            // 512
  float* perp = out + 2359808;            // 1
  float* oidx = out + 2359809;            // 512 (as float)

  float* ws    = (float*)d_ws;
  float* x     = ws;                      // [MROWS,128]
  float* hbf   = ws + 2359296;            // [MROWS,128]
  float* raw   = ws + 4718592;            // [MROWS,128]
  float* yo    = ws + 7077888;            // [MROWS,64]
  float* wt    = ws + 8257536;            // [9,64,128,2]
  float* sum   = ws + 8404992;            // [128]
  float* ssq   = ws + 8405120;            // [128]
  float* flat  = ws + 8405248;            // [512,128]
  float* flatN = ws + 8470784;            // [512,128]
  float* fmean = ws + 8536320;            // [128]
  float* fvar  = ws + 8536448;            // [128]
  float* ksq   = ws + 8536576;            // [512]
  float* cnts  = ws + 8537088;            // [512]
  int*   idxb  = (int*)(ws + 8537600);    // [512]

  // ---- stem: x = relu(conv(s0, W) + conv(s1, Ws)) ----
  k_nchw2nhwc<<<9216, 256, 0, stream>>>(s0, hbf);
  k_nchw2nhwc<<<9216, 256, 0, stream>>>(s1, raw);
  k_transpose_wp<<<576, 256, 0, stream>>>(conv_w, wt, 128);
  k_conv3_wmma<128, 0><<<576, 256, 0, stream>>>(hbf, wt, x);
  k_transpose_wp<<<576, 256, 0, stream>>>(convs_w, wt, 128);
  k_conv3_wmma<128, 3><<<576, 256, 0, stream>>>(raw, wt, x);   // acc + relu

  // ---- residual tower ----
  for (int i = 0; i < 6; ++i) {
    k_transpose_wp<<<576, 256, 0, stream>>>(rb_c1_w + (size_t)i * 147456, wt, 128);
    k_conv3_wmma<128, 0><<<576, 256, 0, stream>>>(x, wt, raw);
    k_zero<<<1, 256, 0, stream>>>(sum, 256);                   // sum+ssq contiguous
    k_chan_stats<<<144, 256, 0, stream>>>(raw, sum, ssq);
    k_bn_act<<<9216, 256, 0, stream>>>(raw, sum, ssq,
        rb_bn1_s + i * 128, rb_bn1_b + i * 128, nullptr, hbf);
    k_transpose_wp<<<576, 256, 0, stream>>>(rb_c2_w + (size_t)i * 147456, wt, 128);
    k_conv3_wmma<128, 0><<<576, 256, 0, stream>>>(hbf, wt, raw);
    k_zero<<<1, 256, 0, stream>>>(sum, 256);
    k_chan_stats<<<144, 256, 0, stream>>>(raw, sum, ssq);
    k_bn_act<<<9216, 256, 0, stream>>>(raw, sum, ssq,
        rb_bn2_s + i * 128, rb_bn2_b + i * 128, x, x);          // + resid, relu
  }

  // ---- conv_out (128 -> 64) ----
  k_transpose_wp<<<288, 256, 0, stream>>>(conv_out_w, wt, 64);
  k_conv3_wmma<64, 0><<<576, 128, 0, stream>>>(x, wt, yo);     // 4 waves/WG

  // ---- FC + BN1d ----
  k_fc_wmma<<<32, 256, 0, stream>>>(yo, fc_w, fc_b, flat);
  k_feat_stats<<<128, 256, 0, stream>>>(flat, fmean, fvar);
  k_bn1d<<<256, 256, 0, stream>>>(flat, fmean, fvar, flatN);

  // ---- VQ ----
  k_emb_sq<<<2, 256, 0, stream>>>(emb, ksq);
  k_zero<<<2, 256, 0, stream>>>(cnts, 512);
  k_vq_argmin<<<512, 256, 0, stream>>>(flatN, emb, ksq, idxb, cnts, oidx);
  k_vq_out<<<512, 128, 0, stream>>>(flatN, emb, idxb, z, loss);
  k_perp<<<1, 512, 0, stream>>>(cnts, perp);
}